// cg_interaction_29480655519978
// MI455X (gfx1250) — compile-verified
//
#include <hip/hip_runtime.h>

// ---------------------------------------------------------------------------
// Fused FullyConnectedTensorProduct (0e+1o+2e)^2 -> 0e+1o+2e  +  two Linear(9,9)
//   out[e,c,k] = sum_ij C[i,j,k] x[e,c,i] y[e,c,j]  +  [x|y] @ Wcat + (bfx+bfy)
// Sparse CG bilinear (83 nonzeros) in VALU; dense 18->9 linear on the matrix
// unit via V_WMMA_F32_16X16X4_F32; staging via GLOBAL_LOAD_ASYNC_TO_LDS_B128 /
// GLOBAL_STORE_ASYNC_FROM_LDS_B128 (ASYNCcnt) when the toolchain exposes them.
// ---------------------------------------------------------------------------

typedef float v2f __attribute__((ext_vector_type(2)));
typedef float v4f __attribute__((ext_vector_type(4)));
typedef float v8f __attribute__((ext_vector_type(8)));
typedef int   v4i __attribute__((ext_vector_type(4)));

// async-builtin pointer types: 16-byte int vectors in explicit address spaces
typedef __attribute__((address_space(1))) v4i g_v4i;   // global
typedef __attribute__((address_space(3))) v4i l_v4i;   // LDS

#if defined(__has_builtin)
#if __has_builtin(__builtin_amdgcn_global_load_async_to_lds_b128) && \
    __has_builtin(__builtin_amdgcn_global_store_async_from_lds_b128)
#define HAVE_ASYNC_LDS 1
#endif
#endif

#define TP_DIM      9
#define NPATH       11
#define NTERM       83
#define PAIRS_PER_BLOCK 256

struct Term { unsigned char i, j, k, p; float v; };

// Real-basis Wigner-3j nonzeros, unit-norm per block, times sqrt((2c+1)/fan_in)
// (fan_in: c=0 ->3, c=1 ->4, c=2 ->4). Index layout: l=0 ->[0], l=1 ->[1..3],
// l=2 ->[4..8]. Path order matches the reference PATHS list.
static constexpr Term TERMS[NTERM] = {
    // path 0: (0,0,0)
    {0,0,0, 0, 0.57735027f},
    // path 1: (0,1,1)
    {0,1,1, 1, 0.5f}, {0,2,2, 1, 0.5f}, {0,3,3, 1, 0.5f},
    // path 2: (0,2,2)
    {0,4,4, 2, 0.5f}, {0,5,5, 2, 0.5f}, {0,6,6, 2, 0.5f}, {0,7,7, 2, 0.5f}, {0,8,8, 2, 0.5f},
    // path 3: (1,0,1)
    {1,0,1, 3, 0.5f}, {2,0,2, 3, 0.5f}, {3,0,3, 3, 0.5f},
    // path 4: (1,1,0)
    {1,1,0, 4, 0.33333333f}, {2,2,0, 4, 0.33333333f}, {3,3,0, 4, 0.33333333f},
    // path 5: (1,1,2)
    {3,1,4, 5, 0.35355339f}, {1,3,4, 5, 0.35355339f},
    {1,2,5, 5, 0.35355339f}, {2,1,5, 5, 0.35355339f},
    {1,1,6, 5,-0.20412415f}, {2,2,6, 5, 0.40824829f}, {3,3,6, 5,-0.20412415f},
    {2,3,7, 5, 0.35355339f}, {3,2,7, 5, 0.35355339f},
    {3,3,8, 5, 0.35355339f}, {1,1,8, 5,-0.35355339f},
    // path 6: (1,2,1)
    {3,4,1, 6, 0.27386128f}, {1,4,3, 6, 0.27386128f},
    {1,5,2, 6, 0.27386128f}, {2,5,1, 6, 0.27386128f},
    {1,6,1, 6,-0.15811388f}, {2,6,2, 6, 0.31622777f}, {3,6,3, 6,-0.15811388f},
    {2,7,3, 6, 0.27386128f}, {3,7,2, 6, 0.27386128f},
    {3,8,3, 6, 0.27386128f}, {1,8,1, 6,-0.27386128f},
    // path 7: (2,0,2)
    {4,0,4, 7, 0.5f}, {5,0,5, 7, 0.5f}, {6,0,6, 7, 0.5f}, {7,0,7, 7, 0.5f}, {8,0,8, 7, 0.5f},
    // path 8: (2,1,1)
    {4,3,1, 8, 0.27386128f}, {4,1,3, 8, 0.27386128f},
    {5,1,2, 8, 0.27386128f}, {5,2,1, 8, 0.27386128f},
    {6,1,1, 8,-0.15811388f}, {6,2,2, 8, 0.31622777f}, {6,3,3, 8,-0.15811388f},
    {7,2,3, 8, 0.27386128f}, {7,3,2, 8, 0.27386128f},
    {8,3,3, 8, 0.27386128f}, {8,1,1, 8,-0.27386128f},
    // path 9: (2,2,0)
    {4,4,0, 9, 0.25819889f}, {5,5,0, 9, 0.25819889f}, {6,6,0, 9, 0.25819889f},
    {7,7,0, 9, 0.25819889f}, {8,8,0, 9, 0.25819889f},
    // path 10: (2,2,2)
    {4,4,6,10,-0.26726124f}, {4,6,4,10,-0.26726124f}, {6,4,4,10,-0.26726124f},
    {8,8,6,10,-0.26726124f}, {8,6,8,10,-0.26726124f}, {6,8,8,10,-0.26726124f},
    {6,6,6,10, 0.26726124f},
    {5,5,6,10, 0.13363062f}, {5,6,5,10, 0.13363062f}, {6,5,5,10, 0.13363062f},
    {7,7,6,10, 0.13363062f}, {7,6,7,10, 0.13363062f}, {6,7,7,10, 0.13363062f},
    {5,5,8,10,-0.23145502f}, {5,8,5,10,-0.23145502f}, {8,5,5,10,-0.23145502f},
    {7,7,8,10, 0.23145502f}, {7,8,7,10, 0.23145502f}, {8,7,7,10, 0.23145502f},
    {4,5,7,10, 0.23145502f}, {4,7,5,10, 0.23145502f}, {5,4,7,10, 0.23145502f},
    {5,7,4,10, 0.23145502f}, {7,4,5,10, 0.23145502f}, {7,5,4,10, 0.23145502f},
};

__global__ __launch_bounds__(256) void ftp_kernel(
    const float* __restrict__ xg, const float* __restrict__ yg,
    const float* __restrict__ tpw,
    const float* __restrict__ Wfx, const float* __restrict__ bfx,
    const float* __restrict__ Wfy, const float* __restrict__ bfy,
    float* __restrict__ outg)
{
    // packed [pair][9] images: exact copies of the global chunks (stride 9 is
    // coprime with the 32 LDS banks -> per-thread row access is conflict-free)
    __shared__ alignas(16) float xs[PAIRS_PER_BLOCK * TP_DIM];
    __shared__ alignas(16) float ys[PAIRS_PER_BLOCK * TP_DIM];
    __shared__ alignas(16) float os[PAIRS_PER_BLOCK * TP_DIM];

    const int tid = threadIdx.x;
    const long long gbase = (long long)blockIdx.x * (PAIRS_PER_BLOCK * TP_DIM);

    // ---- stage x,y -> LDS (2304 floats each, 576 x b128) ----
#ifdef HAVE_ASYNC_LDS
#pragma unroll
    for (int it = 0; it < 3; ++it) {
        int idx4 = tid + it * 256;
        if (idx4 < 576) {
            __builtin_amdgcn_global_load_async_to_lds_b128(
                (g_v4i*)(xg + gbase) + idx4, (l_v4i*)xs + idx4, 0, 0);
            __builtin_amdgcn_global_load_async_to_lds_b128(
                (g_v4i*)(yg + gbase) + idx4, (l_v4i*)ys + idx4, 0, 0);
        }
    }
    asm volatile("s_wait_asynccnt 0x0" ::: "memory");
#else
    const v4f* x4 = (const v4f*)(xg + gbase);
    const v4f* y4 = (const v4f*)(yg + gbase);
#pragma unroll
    for (int it = 0; it < 3; ++it) {
        int idx4 = tid + it * 256;
        if (idx4 < 576) {
            ((v4f*)xs)[idx4] = __builtin_nontemporal_load(&x4[idx4]);
            ((v4f*)ys)[idx4] = __builtin_nontemporal_load(&y4[idx4]);
        }
    }
#endif

    // ---- build B (weight) fragments: Wcat[k][n], k=0..17 -> Wfx^T|Wfy^T ----
    // B layout (4x16, f32): v0: lanes0-15 K=0 / lanes16-31 K=2; v1: K=1 / K=3
    const int lane = tid & 31;
    const int lh   = lane >> 4;     // lane-half selector
    const int ln   = lane & 15;     // N (output column) / M within tile
    v2f Bf[5];
#pragma unroll
    for (int t = 0; t < 5; ++t) {
        int k0 = 4 * t + 2 * lh;
#pragma unroll
        for (int e = 0; e < 2; ++e) {
            int k = k0 + e;
            float wv = 0.0f;
            if (ln < TP_DIM) {
                if (k < TP_DIM)          wv = Wfx[ln * TP_DIM + k];             // x @ Wfx^T
                else if (k < 2 * TP_DIM) wv = Wfy[ln * TP_DIM + (k - TP_DIM)];  // y @ Wfy^T
            }
            Bf[t][e] = wv;
        }
    }
    const float bias = (ln < TP_DIM) ? (bfx[ln] + bfy[ln]) : 0.0f;

    __syncthreads();

    // ---- linear branch on the matrix unit: D = A(16x18) * Wcat(18x9) + bias ----
    // A layout (16x4, f32): lane L = row M (L&15); v0/v1 hold K = (0,1)|(2,3)
    const int wave = tid >> 5;
#pragma unroll
    for (int tile = 0; tile < 2; ++tile) {
        const int pbase = wave * 32 + tile * 16;
        const float* xrow = &xs[(pbase + ln) * TP_DIM];
        const float* yrow = &ys[(pbase + ln) * TP_DIM];
        v8f acc = {bias, bias, bias, bias, bias, bias, bias, bias};

        v2f a0; a0.x = xrow[0 + 2 * lh]; a0.y = xrow[1 + 2 * lh];      // K 0..3
        acc = __builtin_amdgcn_wmma_f32_16x16x4_f32(false, a0, false, Bf[0], (short)0, acc, false, false);

        v2f a1; a1.x = xrow[4 + 2 * lh]; a1.y = xrow[5 + 2 * lh];      // K 4..7
        acc = __builtin_amdgcn_wmma_f32_16x16x4_f32(false, a1, false, Bf[1], (short)0, acc, false, false);

        v2f a2;                                                         // K 8..11
        a2.x = lh ? yrow[1] : xrow[8];
        a2.y = lh ? yrow[2] : yrow[0];
        acc = __builtin_amdgcn_wmma_f32_16x16x4_f32(false, a2, false, Bf[2], (short)0, acc, false, false);

        v2f a3; a3.x = yrow[3 + 2 * lh]; a3.y = yrow[4 + 2 * lh];      // K 12..15
        acc = __builtin_amdgcn_wmma_f32_16x16x4_f32(false, a3, false, Bf[3], (short)0, acc, false, false);

        v2f a4;                                                         // K 16..19 (18,19 = pad)
        a4.x = lh ? 0.0f : yrow[7];
        a4.y = lh ? 0.0f : yrow[8];
        acc = __builtin_amdgcn_wmma_f32_16x16x4_f32(false, a4, false, Bf[4], (short)0, acc, false, false);

        // D layout: VGPR r, lanes0-15 -> (M=r, N=lane); lanes16-31 -> (M=8+r).
        // Only columns n<9 are real outputs; mask so we don't clobber packed rows.
        if (ln < TP_DIM) {
#pragma unroll
            for (int r = 0; r < 8; ++r)
                os[(pbase + r + 8 * lh) * TP_DIM + ln] = acc[r];
        }
    }
    __syncthreads();

    // ---- per-pair sparse CG bilinear, fused with linear result ----
    float xr[TP_DIM], yr[TP_DIM], o[TP_DIM], w[NPATH];
#pragma unroll
    for (int f = 0; f < TP_DIM; ++f) {
        xr[f] = xs[tid * TP_DIM + f];
        yr[f] = ys[tid * TP_DIM + f];
    }
#pragma unroll
    for (int p = 0; p < NPATH; ++p) w[p] = tpw[p];
#pragma unroll
    for (int k = 0; k < TP_DIM; ++k) o[k] = os[tid * TP_DIM + k];

#pragma unroll
    for (int t = 0; t < NTERM; ++t) {
        o[TERMS[t].k] = fmaf(TERMS[t].v * w[TERMS[t].p],
                             xr[TERMS[t].i] * yr[TERMS[t].j],
                             o[TERMS[t].k]);
    }

    // own-row read/write only -> no extra barrier needed before the overwrite
#pragma unroll
    for (int k = 0; k < TP_DIM; ++k) os[tid * TP_DIM + k] = o[k];
    __syncthreads();

    // ---- store the 2304-float chunk back (contiguous, 576 x b128) ----
#ifdef HAVE_ASYNC_LDS
#pragma unroll
    for (int it = 0; it < 3; ++it) {
        int idx4 = tid + it * 256;
        if (idx4 < 576) {
            __builtin_amdgcn_global_store_async_from_lds_b128(
                (g_v4i*)(outg + gbase) + idx4, (l_v4i*)os + idx4, 0, 0);
        }
    }
    // S_ENDPGM performs an implicit wait-idle (all counters, incl. ASYNCcnt)
#else
    v4f* o4 = (v4f*)(outg + gbase);
#pragma unroll
    for (int it = 0; it < 3; ++it) {
        int idx4 = tid + it * 256;
        if (idx4 < 576)
            __builtin_nontemporal_store(((const v4f*)os)[idx4], &o4[idx4]);
    }
#endif
}

extern "C" void kernel_launch(void* const* d_in, const int* in_sizes, int n_in,
                              void* d_out, int out_size, void* d_ws, size_t ws_size,
                              hipStream_t stream) {
    const float* x   = (const float*)d_in[0];
    const float* y   = (const float*)d_in[1];
    const float* tpw = (const float*)d_in[2];
    const float* Wfx = (const float*)d_in[3];
    const float* bfx = (const float*)d_in[4];
    const float* Wfy = (const float*)d_in[5];
    const float* bfy = (const float*)d_in[6];
    float* out = (float*)d_out;

    const long long pairs = (long long)in_sizes[0] / TP_DIM;   // E*CH = 16,777,216
    const int blocks = (int)(pairs / PAIRS_PER_BLOCK);          // 65,536
    ftp_kernel<<<blocks, 256, 0, stream>>>(x, y, tpw, Wfx, bfx, Wfy, bfy, out);
}